// PGRA_25658134626706
// MI455X (gfx1250) — compile-verified
//
#include <hip/hip_runtime.h>
#include <math.h>

#define EMBV 128
#define KNB  16
#define NREL 32
#define EPSV 1e-8f

typedef __attribute__((ext_vector_type(2))) float v2f;
typedef __attribute__((ext_vector_type(8))) float v8f;

// ---------------------------------------------------------------------------
// Kernel 1: precompute S[rel][r] = |dot(rela[rel], rela[r])| /
//                                  (||rela[rel]|| * ||rela[r]|| + EPS)
// Only 32x32 = 1024 entries; one block, one thread per entry.
// ---------------------------------------------------------------------------
__global__ __launch_bounds__(1024) void pgra_scores(const float* __restrict__ rela,
                                                    float* __restrict__ S) {
    int t = threadIdx.x;
    int i = t >> 5;        // target relation
    int j = t & 31;        // neighbor relation
    const float* a = rela + i * EMBV;
    const float* b = rela + j * EMBV;
    float dab = 0.f, daa = 0.f, dbb = 0.f;
    #pragma unroll 4
    for (int e = 0; e < EMBV; ++e) {
        float x = a[e], y = b[e];
        dab += x * y; daa += x * x; dbb += y * y;
    }
    S[i * NREL + j] = fabsf(dab) / (sqrtf(daa) * sqrtf(dbb) + EPSV);
}

// ---------------------------------------------------------------------------
// Kernel 2: stage-A attention aggregate. One wave per (batch, group) pair.
//   gid = b*16 + g ; e1 = adj_node[node[b]*16 + g]
//   att = softmax_k( S[rel][ adj_rela[e1*16+k] ] )
//   AGG[gid] = proj[rel] ⊙ Σ_k att_k * node_table[ adj_node[e1*16+k] ]
// Each lane handles 4 of the 128 emb floats (float4) -> fully coalesced
// 512B row reads; index/score loads are wave-uniform (single cacheline).
// ---------------------------------------------------------------------------
__global__ __launch_bounds__(256) void pgra_stageA(const int* __restrict__ node,
                                                   const int* __restrict__ relation,
                                                   const int* __restrict__ adj_node,
                                                   const int* __restrict__ adj_rela,
                                                   const float* __restrict__ node_table,
                                                   const float* __restrict__ proj_table,
                                                   const float* __restrict__ S,
                                                   float* __restrict__ AGG) {
    int lane = threadIdx.x & 31;
    int gid  = blockIdx.x * 8 + (threadIdx.x >> 5);   // 8 waves per block
    int b = gid >> 4;
    int g = gid & 15;
    int rel = relation[b];
    int e1  = adj_node[(size_t)node[b] * KNB + g];
    const int* an = adj_node + (size_t)e1 * KNB;
    const int* ar = adj_rela + (size_t)e1 * KNB;

    float s[KNB];
    float m = -1e30f;
    #pragma unroll
    for (int k = 0; k < KNB; ++k) { s[k] = S[rel * NREL + ar[k]]; m = fmaxf(m, s[k]); }
    float sum = 0.f;
    #pragma unroll
    for (int k = 0; k < KNB; ++k) { s[k] = expf(s[k] - m); sum += s[k]; }
    float inv = 1.f / sum;

    float4 acc = make_float4(0.f, 0.f, 0.f, 0.f);
    #pragma unroll
    for (int k = 0; k < KNB; ++k) {
        const float4* row = (const float4*)(node_table + (size_t)an[k] * EMBV);
        float4 v = row[lane];
        float  w = s[k] * inv;
        acc.x += w * v.x; acc.y += w * v.y; acc.z += w * v.z; acc.w += w * v.w;
    }
    float4 pv = ((const float4*)(proj_table + (size_t)rel * EMBV))[lane];
    acc.x *= pv.x; acc.y *= pv.y; acc.z *= pv.z; acc.w *= pv.w;
    ((float4*)(AGG + (size_t)gid * EMBV))[lane] = acc;
}

// ---------------------------------------------------------------------------
// Kernel 3/5: D = tanh(A @ W + bias) via V_WMMA_F32_16X16X4_F32.
// A: (M x 128) row-major, W: (128 x 128) row-major, D: (M x 128).
// Block = 256 threads = 8 waves; block covers 16 rows x 128 cols,
// wave w owns the 16x16 tile at cols [16w, 16w+16). K loop: 128/4 = 32 WMMAs.
// f32 WMMA fragment layout (ISA 7.12.2):
//   A 16x4 : lane<16 -> K={kb,kb+1}, lane>=16 -> K={kb+2,kb+3}, M = lane&15
//   B 4x16 : same K split, N = lane&15
//   C 16x16: VGPR j -> row (j + 8*(lane>=16)), col lane&15
// ---------------------------------------------------------------------------
__global__ __launch_bounds__(256) void pgra_gemm_tanh(const float* __restrict__ A,
                                                      const float* __restrict__ W,
                                                      const float* __restrict__ bias,
                                                      float* __restrict__ D) {
    int lane  = threadIdx.x & 31;
    int wv    = threadIdx.x >> 5;        // 0..7 -> N tile
    int mBase = blockIdx.x * 16;
    int nBase = wv * 16;
    int row   = lane & 15;
    int hi    = lane >> 4;               // 0 or 1
    int klo   = hi * 2;

    v8f c = {0.f, 0.f, 0.f, 0.f, 0.f, 0.f, 0.f, 0.f};
    const float* arow = A + (size_t)(mBase + row) * EMBV + klo;
    #pragma unroll 4
    for (int kb = 0; kb < EMBV; kb += 4) {
        v2f a;
        a.x = arow[kb];
        a.y = arow[kb + 1];
        v2f bb;
        bb.x = W[(size_t)(kb + klo) * EMBV + nBase + row];
        bb.y = W[(size_t)(kb + klo + 1) * EMBV + nBase + row];
        c = __builtin_amdgcn_wmma_f32_16x16x4_f32(false, a, false, bb,
                                                  (short)0, c, false, false);
    }

    int n = nBase + row;
    float bn = bias[n];
    #pragma unroll
    for (int j = 0; j < 8; ++j) {
        int mrow = mBase + j + hi * 8;
        D[(size_t)mrow * EMBV + n] = tanhf(c[j] + bn);
    }
}

// ---------------------------------------------------------------------------
// Kernel 4: stage-B attention aggregate. One wave per batch row.
//   att = softmax_k( S[rel][ adj_rela[node[b]*16+k] ] )
//   OUT2[b] = Σ_k att_k * H1[b*16+k]
// ---------------------------------------------------------------------------
__global__ __launch_bounds__(256) void pgra_stageB(const int* __restrict__ node,
                                                   const int* __restrict__ relation,
                                                   const int* __restrict__ adj_rela,
                                                   const float* __restrict__ H1,
                                                   const float* __restrict__ S,
                                                   float* __restrict__ OUT2) {
    int lane = threadIdx.x & 31;
    int b    = blockIdx.x * 8 + (threadIdx.x >> 5);
    int rel  = relation[b];
    const int* ar = adj_rela + (size_t)node[b] * KNB;

    float s[KNB];
    float m = -1e30f;
    #pragma unroll
    for (int k = 0; k < KNB; ++k) { s[k] = S[rel * NREL + ar[k]]; m = fmaxf(m, s[k]); }
    float sum = 0.f;
    #pragma unroll
    for (int k = 0; k < KNB; ++k) { s[k] = expf(s[k] - m); sum += s[k]; }
    float inv = 1.f / sum;

    float4 acc = make_float4(0.f, 0.f, 0.f, 0.f);
    #pragma unroll
    for (int k = 0; k < KNB; ++k) {
        const float4* row = (const float4*)(H1 + ((size_t)b * KNB + k) * EMBV);
        float4 v = row[lane];
        float  w = s[k] * inv;
        acc.x += w * v.x; acc.y += w * v.y; acc.z += w * v.z; acc.w += w * v.w;
    }
    ((float4*)(OUT2 + (size_t)b * EMBV))[lane] = acc;
}

// ---------------------------------------------------------------------------
// Launch
// ---------------------------------------------------------------------------
extern "C" void kernel_launch(void* const* d_in, const int* in_sizes, int n_in,
                              void* d_out, int out_size, void* d_ws, size_t ws_size,
                              hipStream_t stream) {
    (void)n_in; (void)out_size; (void)ws_size;
    const int*   node       = (const int*)d_in[0];
    const int*   relation   = (const int*)d_in[1];
    const int*   adj_node   = (const int*)d_in[2];
    const int*   adj_rela   = (const int*)d_in[3];
    const float* node_table = (const float*)d_in[4];
    const float* rela_table = (const float*)d_in[5];
    const float* proj_table = (const float*)d_in[6];
    const float* W0         = (const float*)d_in[7];
    const float* b0         = (const float*)d_in[8];
    const float* W1         = (const float*)d_in[9];
    const float* b1         = (const float*)d_in[10];
    float*       out        = (float*)d_out;

    const int B = in_sizes[0];            // 2048

    float* ws   = (float*)d_ws;
    float* S    = ws;                                  // 32*32
    float* AGG  = ws + 1024;                           // B*16*128
    float* H1   = AGG + (size_t)B * KNB * EMBV;        // B*16*128
    float* OUT2 = H1  + (size_t)B * KNB * EMBV;        // B*128

    // 1) 32x32 abs-cos score table
    pgra_scores<<<1, 1024, 0, stream>>>(rela_table, S);
    // 2) second-hop attention aggregate -> AGG (B*16 rows)
    pgra_stageA<<<(B * KNB) / 8, 256, 0, stream>>>(node, relation, adj_node, adj_rela,
                                                   node_table, proj_table, S, AGG);
    // 3) H1 = tanh(AGG @ W0 + b0)   (M = B*16)
    pgra_gemm_tanh<<<(B * KNB) / 16, 256, 0, stream>>>(AGG, W0, b0, H1);
    // 4) first-hop attention aggregate over H1 -> OUT2 (B rows)
    pgra_stageB<<<B / 8, 256, 0, stream>>>(node, relation, adj_rela, H1, S, OUT2);
    // 5) out = tanh(OUT2 @ W1 + b1)  (M = B)
    pgra_gemm_tanh<<<B / 16, 256, 0, stream>>>(OUT2, W1, b1, out);
}